// PositionalEncoding_25288767438916
// MI455X (gfx1250) — compile-verified
//
#include <hip/hip_runtime.h>

typedef __attribute__((ext_vector_type(2))) float v2f;
typedef __attribute__((ext_vector_type(8))) float v8f;

#define D_MODEL 1024
#define SEQ     256
#define BATCH   256

// One wave computes one 16(b) x 16(d) output tile at fixed s via a single
// V_WMMA_F32_16X16X4_F32:  D(16x16) = A(16x4) * B(4x16), C = 0.
//   A rows (M = batch within tile): { x0, x1, x2, 1.0 }
//   B rows (N = d within tile):     { W_xy[:,0], W_xy[:,1], W_seg, b_xy+b_seg+pe[s,:] }
// 8 waves per block cover 8 consecutive d-tiles (reusing the same A from cache).
__global__ __launch_bounds__(256)
void pe_wmma_kernel(const float* __restrict__ x,      // [S,B,3]
                    const float* __restrict__ W_xy,   // [D,2]
                    const float* __restrict__ b_xy,   // [D]
                    const float* __restrict__ W_seg,  // [D,1]
                    const float* __restrict__ b_seg,  // [D]
                    const float* __restrict__ pe,     // [L,1,D]
                    float* __restrict__ out)          // [S,B,D]
{
    const int lane = threadIdx.x & 31;
    const int wave = threadIdx.x >> 5;
    const int half = lane >> 4;      // 0: lanes 0-15, 1: lanes 16-31
    const int l16  = lane & 15;

    const int s  = blockIdx.z;
    const int b0 = blockIdx.y << 4;
    const int d0 = ((blockIdx.x << 3) + wave) << 4;

    // ---- A matrix 16x4 f32: lane l -> M = l%16, K = half*2 + vgpr ----
    // K = 0,1,2 come from x[s, b0+M, 0..2]; K = 3 is the constant 1.0
    const float* xr = x + (size_t)(s * BATCH + b0 + l16) * 3;
    v2f a;
    a.x = xr[half * 2];                 // K = 0 (half=0) or K = 2 (half=1)
    a.y = half ? 1.0f : xr[1];          // K = 1 (half=0) or K = 3 -> 1.0
    // (xr[1] is in-bounds for every lane; the select lowers to v_cndmask,
    //  so EXEC stays all-1s as WMMA requires.)

    // ---- B matrix 4x16 f32: lane l -> N = l%16, K = half*2 + vgpr ----
    const int d = d0 + l16;
    const float wx0  = W_xy[2 * d];                                 // K=0 row
    const float wx1  = W_xy[2 * d + 1];                             // K=1 row
    const float wsg  = W_seg[d];                                    // K=2 row
    const float bias = b_xy[d] + b_seg[d] + pe[s * D_MODEL + d];    // K=3 row
    v2f bm;
    bm.x = half ? wsg  : wx0;
    bm.y = half ? bias : wx1;

    // ---- D = A*B + 0 ----
    v8f c = {};
    c = __builtin_amdgcn_wmma_f32_16x16x4_f32(
            /*neg_a=*/false, a, /*neg_b=*/false, bm,
            /*c_mod=*/(short)0, c, /*reuse_a=*/false, /*reuse_b=*/false);

    // ---- Store: VGPR r holds row M = half*8 + r, col N = l%16 ----
    float* o = out + (size_t)(s * BATCH + b0 + half * 8) * D_MODEL + d;
#pragma unroll
    for (int r = 0; r < 8; ++r)
        o[r * D_MODEL] = c[r];
}

extern "C" void kernel_launch(void* const* d_in, const int* in_sizes, int n_in,
                              void* d_out, int out_size, void* d_ws, size_t ws_size,
                              hipStream_t stream) {
    const float* x     = (const float*)d_in[0];
    const float* W_xy  = (const float*)d_in[1];
    const float* b_xy  = (const float*)d_in[2];
    const float* W_seg = (const float*)d_in[3];
    const float* b_seg = (const float*)d_in[4];
    const float* pe    = (const float*)d_in[5];
    float* out = (float*)d_out;

    // 8 waves/block, each one 16x16 tile: grid = (D/128, B/16, S)
    dim3 grid(D_MODEL / 128, BATCH / 16, SEQ);
    pe_wmma_kernel<<<grid, dim3(256), 0, stream>>>(x, W_xy, b_xy, W_seg, b_seg, pe, out);
}